// Net_70153995813538
// MI455X (gfx1250) — compile-verified
//
#include <hip/hip_runtime.h>
#include <math.h>

#define N_NODES  100000
#define N_EDGES  3200000
#define N_GRAPHS 512
#define D_FEAT   128
#define M_TILES  (N_NODES / 16)   // 6250, exact

typedef float v2f __attribute__((ext_vector_type(2)));
typedef float v8f __attribute__((ext_vector_type(8)));

// ---------------------------------------------------------------------------
// GEMM: Y[M,32] = X[M,K] @ W[K,32] using V_WMMA_F32_16X16X4_F32.
// One wave computes a 16x32 output tile (two 16x16 f32 accumulators).
//
// A-frag (16x4 MxK): lanes 0-15 = M rows, VGPR pair = K{0,1}; lanes 16-31
//   hold K{2,3}  -> one 8B-aligned v2f load per K-step.
// B-frag (4x16 KxN): lanes 0-15 = rows K{0,1}, lanes 16-31 = rows K{2,3}.
//   W is staged in LDS PRE-PAIRED along K: Wp[k/2][n] = (W[k][n], W[k+1][n])
//   so each B-frag is a single ds_load_b64 into an even VGPR pair (no movs).
// C/D: VGPR r holds M = r + 8*(lane/16), N = lane%16.
// ---------------------------------------------------------------------------
template <int K>
__global__ void gemm_wmma_n32(const float* __restrict__ X,
                              const float* __restrict__ W,
                              float* __restrict__ Y, int mtiles) {
    __shared__ v2f Wp[(K / 2) * 32];
    for (int i = threadIdx.x; i < (K / 2) * 32; i += blockDim.x) {
        const int kk = i >> 5;          // K-pair index (k = 2*kk)
        const int nn = i & 31;
        v2f t;
        t.x = W[(2 * kk)     * 32 + nn];
        t.y = W[(2 * kk + 1) * 32 + nn];
        Wp[i] = t;
    }
    __syncthreads();

    const int wave = threadIdx.x >> 5;
    const int lane = threadIdx.x & 31;
    const int tile = blockIdx.x * (blockDim.x >> 5) + wave;
    if (tile >= mtiles) return;                 // wave-uniform: EXEC stays all-1s

    const int n     = lane & 15;
    const int lh    = lane >> 4;                // half-wave select (K pair 0 / 1)
    const int khalf = lh * 2;
    const float* __restrict__ xrow = X + (size_t)(tile * 16 + n) * K;  // m = lane%16

    v8f c0 = {}; v8f c1 = {};
    for (int kb = 0; kb < K; kb += 4) {
        const v2f a  = *(const v2f*)(xrow + kb + khalf);       // global_load_b64
        const v2f b0 = Wp[(kb / 2 + lh) * 32 + n];             // ds_load_b64
        const v2f b1 = Wp[(kb / 2 + lh) * 32 + 16 + n];        // ds_load_b64
        c0 = __builtin_amdgcn_wmma_f32_16x16x4_f32(false, a, false, b0, (short)0, c0, false, false);
        c1 = __builtin_amdgcn_wmma_f32_16x16x4_f32(false, a, false, b1, (short)0, c1, false, false);
    }

    float* __restrict__ yb = Y + (size_t)(tile * 16 + lh * 8) * 32;
#pragma unroll
    for (int r = 0; r < 8; ++r) {
        yb[r * 32 + n]      = c0[r];
        yb[r * 32 + 16 + n] = c1[r];
    }
}

// ---------------------------------------------------------------------------
// SpMM scatter: acc[row[e], :] += w[e] * h[col[e], :]. One edge per wave;
// lane = feature. 128B coalesced gather per edge; f32 atomics resolve in the
// 192MB L2 (h and acc are 12.8 MB each -> fully L2-resident).
// ---------------------------------------------------------------------------
__global__ void spmm_f32(const int* __restrict__ rows, const int* __restrict__ cols,
                         const float* __restrict__ wgt, const float* __restrict__ h,
                         float* __restrict__ acc, int nE) {
    const int e = blockIdx.x * (blockDim.x >> 5) + (threadIdx.x >> 5);
    const int f = threadIdx.x & 31;
    if (e >= nE) return;
    const int   r = rows[e];
    const int   c = cols[e];
    const float w = wgt[e];
    atomicAdd(&acc[(size_t)r * 32 + f], w * h[(size_t)c * 32 + f]);
}

__global__ void zero_f32(float* __restrict__ p, int n) {
    int i = blockIdx.x * blockDim.x + threadIdx.x;
    if (i < n) p[i] = 0.0f;
}

__global__ void bias_elu(const float* __restrict__ acc, const float* __restrict__ b,
                         float* __restrict__ out, int n) {
    int i = blockIdx.x * blockDim.x + threadIdx.x;
    if (i < n) {
        float v = acc[i] + b[i & 31];
        out[i] = (v > 0.0f) ? v : expm1f(v);
    }
}

__global__ void seg_pool(const float* __restrict__ h, const int* __restrict__ seg,
                         float* __restrict__ g, int n) {
    int i = blockIdx.x * blockDim.x + threadIdx.x;
    if (i < n) {
        int node = i >> 5, f = i & 31;
        atomicAdd(&g[(size_t)seg[node] * 32 + f], h[i]);
    }
}

// Tiny MLP head: relu(g@Wd1+bd1) -> relu(@Wd2+bd2) -> sigmoid(@Wd3+bd3)
__global__ void mlp_head(const float* __restrict__ g,
                         const float* __restrict__ Wd1, const float* __restrict__ bd1,
                         const float* __restrict__ Wd2, const float* __restrict__ bd2,
                         const float* __restrict__ Wd3, const float* __restrict__ bd3,
                         float* __restrict__ out) {
    int t = blockIdx.x * blockDim.x + threadIdx.x;
    if (t >= N_GRAPHS) return;
    float gin[32];
#pragma unroll
    for (int k = 0; k < 32; ++k) gin[k] = g[t * 32 + k];
    float h1[64];
    for (int j = 0; j < 64; ++j) {
        float s = bd1[j];
        for (int k = 0; k < 32; ++k) s = fmaf(gin[k], Wd1[k * 64 + j], s);
        h1[j] = fmaxf(s, 0.0f);
    }
    float h2[32];
    for (int j = 0; j < 32; ++j) {
        float s = bd2[j];
        for (int k = 0; k < 64; ++k) s = fmaf(h1[k], Wd2[k * 32 + j], s);
        h2[j] = fmaxf(s, 0.0f);
    }
    float o = bd3[0];
    for (int k = 0; k < 32; ++k) o = fmaf(h2[k], Wd3[k], o);
    out[t] = 1.0f / (1.0f + __expf(-o));
}

extern "C" void kernel_launch(void* const* d_in, const int* in_sizes, int n_in,
                              void* d_out, int out_size, void* d_ws, size_t ws_size,
                              hipStream_t stream) {
    const float* x    = (const float*)d_in[0];
    const int*   rows = (const int*)d_in[1];
    const int*   cols = rows + N_EDGES;
    const float* wgt  = (const float*)d_in[2];
    const int*   seg  = (const int*)d_in[3];
    const float* W1   = (const float*)d_in[4];
    const float* b1   = (const float*)d_in[5];
    const float* W2   = (const float*)d_in[6];
    const float* b2   = (const float*)d_in[7];
    const float* W3   = (const float*)d_in[8];
    const float* b3   = (const float*)d_in[9];
    const float* Wd1  = (const float*)d_in[10];
    const float* bd1  = (const float*)d_in[11];
    const float* Wd2  = (const float*)d_in[12];
    const float* bd2  = (const float*)d_in[13];
    const float* Wd3  = (const float*)d_in[14];
    const float* bd3  = (const float*)d_in[15];
    float* out = (float*)d_out;

    const int NF = N_NODES * 32;               // 3.2M elements per node buffer
    float* hpre = (float*)d_ws;                // GEMM output / SpMM gather source
    float* acc  = hpre + NF;                   // SpMM accumulator
    float* hcur = acc + NF;                    // post bias+ELU (next layer input)
    float* g    = hcur + NF;                   // [512, 32] pooled

    const dim3 blk(256);
    const int gemm_blocks = (M_TILES + 7) / 8;            // 8 waves/block
    const int spmm_blocks = (N_EDGES + 7) / 8;            // 8 edges/block
    const int ew_blocks   = (NF + 255) / 256;

    // ---- layer 1 (K = 128) ----
    gemm_wmma_n32<128><<<gemm_blocks, blk, 0, stream>>>(x, W1, hpre, M_TILES);
    zero_f32<<<ew_blocks, blk, 0, stream>>>(acc, NF);
    spmm_f32<<<spmm_blocks, blk, 0, stream>>>(rows, cols, wgt, hpre, acc, N_EDGES);
    bias_elu<<<ew_blocks, blk, 0, stream>>>(acc, b1, hcur, NF);

    // ---- layer 2 (K = 32) ----
    gemm_wmma_n32<32><<<gemm_blocks, blk, 0, stream>>>(hcur, W2, hpre, M_TILES);
    zero_f32<<<ew_blocks, blk, 0, stream>>>(acc, NF);
    spmm_f32<<<spmm_blocks, blk, 0, stream>>>(rows, cols, wgt, hpre, acc, N_EDGES);
    bias_elu<<<ew_blocks, blk, 0, stream>>>(acc, b2, hcur, NF);

    // ---- layer 3 (K = 32) ----
    gemm_wmma_n32<32><<<gemm_blocks, blk, 0, stream>>>(hcur, W3, hpre, M_TILES);
    zero_f32<<<ew_blocks, blk, 0, stream>>>(acc, NF);
    spmm_f32<<<spmm_blocks, blk, 0, stream>>>(rows, cols, wgt, hpre, acc, N_EDGES);
    bias_elu<<<ew_blocks, blk, 0, stream>>>(acc, b3, hcur, NF);

    // ---- global sum pool + MLP head ----
    zero_f32<<<(N_GRAPHS * 32 + 255) / 256, blk, 0, stream>>>(g, N_GRAPHS * 32);
    seg_pool<<<ew_blocks, blk, 0, stream>>>(hcur, seg, g, NF);
    mlp_head<<<(N_GRAPHS + 255) / 256, blk, 0, stream>>>(g, Wd1, bd1, Wd2, bd2, Wd3, bd3, out);
}